// ChannelAttention_86517821216050
// MI455X (gfx1250) — compile-verified
//
#include <hip/hip_runtime.h>

// ---------------------------------------------------------------------------
// ChannelAttention for MI455X (gfx1250), fp32 WMMA path (V_WMMA_F32_16X16X4_F32)
//
// B=32, IDF=128, CDF=256, L=64, HW=4096
//   k1: sourceC[b] = W2 @ context[b]                    (4096x256 @ 256x64)
//   k2: attn partials (split-K=8): wC[b] @ sourceC[b]   (128x4096 @ 4096x64)
//   k3: reduce partials + softmax over L; emit attn_sm and transposed attn_c
//   k4: out[b] = attn_sm[b] @ sourceC[b]^T              (128x64 @ 64x4096)
//
// LDS B tiles are stored PAIR-INTERLEAVED: sB2[k/2][n] = {B[k][n], B[k+1][n]},
// so each WMMA B fragment is a single contiguous ds_load_b64 into an adjacent
// (even-aligned) VGPR pair -> no repack v_mov's before v_wmma.
// Each wave register-blocks a 32x32 macro-tile: 2 A + 2 B loads per 4 WMMAs.
// ---------------------------------------------------------------------------

typedef float v2f __attribute__((ext_vector_type(2)));
typedef float v8f __attribute__((ext_vector_type(8)));

#define B_      32
#define IDF_    128
#define CDF_    256
#define LL      64
#define HW_     4096
#define KSPLIT  8
// pair-row strides in float2 units:
//   STRP:  2*STRP  % 64 == 32  -> half-wave b64 loads hit disjoint 32-bank windows
#define STRP    80    // 64-wide tiles (k1/k2): 64 pair-rows * 80 * 8B = 40 KB
#define STRP2   129   // 128-wide transposed tile (k4): conflict-free transpose stores

static __device__ __forceinline__ v8f wmma_f32(v2f a, v2f b, v8f c) {
    // D(16x16,f32) = A(16x4,f32) * B(4x16,f32) + C
    return __builtin_amdgcn_wmma_f32_16x16x4_f32(
        /*neg_a=*/false, a, /*neg_b=*/false, b,
        /*c_mod=*/(short)0, c, /*reuse_a=*/false, /*reuse_b=*/false);
}

static __device__ __forceinline__ void zero8(v8f& v) {
#pragma unroll
    for (int e = 0; e < 8; ++e) v[e] = 0.0f;
}

// Stage a 128(k) x 64(n) row-major global chunk into pair-interleaved LDS.
// 64 pair-rows x 16 col-quads = 1024 units, 4 per thread.
static __device__ __forceinline__ void stage_pairs_128x64(
        float* __restrict__ sB, const float* __restrict__ g, int tid) {
#pragma unroll
    for (int it = 0; it < 4; ++it) {
        int idx = tid + it * 256;          // 0..1023
        int p = idx >> 4;                  // pair-row (k = 2p, 2p+1)
        int c = idx & 15;                  // col quad
        float4 a = *(const float4*)(g + (size_t)(2 * p) * LL + 4 * c);
        float4 b = *(const float4*)(g + (size_t)(2 * p + 1) * LL + 4 * c);
        float* d = &sB[(p * STRP + 4 * c) * 2];
        float4 s0 = {a.x, b.x, a.y, b.y};  // pairs n=4c, 4c+1
        float4 s1 = {a.z, b.z, a.w, b.w};  // pairs n=4c+2, 4c+3
        *(float4*)(d + 0) = s0;
        *(float4*)(d + 4) = s1;
    }
}

// ---------------------------------------------------------------------------
// k1: sourceC[b][o][l] = sum_c W2[o][c] * context[b][c][l]
// grid (HW/128, B), 256 threads; wave (mg,nh) computes rows mg*32..+31,
// cols nh*32..+31 of the 128x64 block.
// ---------------------------------------------------------------------------
__global__ __launch_bounds__(256) void k1_sourceC(const float* __restrict__ W2,
                                                  const float* __restrict__ ctx,
                                                  float* __restrict__ sC) {
    __shared__ float sB[64 * STRP * 2];    // 40 KB
    const int b    = blockIdx.y;
    const int tid  = threadIdx.x;
    const int wave = tid >> 5;
    const int lane = tid & 31;
    const int hf   = lane >> 4;
    const int l16  = lane & 15;
    const int mg   = wave >> 1;
    const int nh   = wave & 1;
    const int m0   = blockIdx.x * 128 + mg * 32;
    const int n0   = nh * 32;

    const float* ctxb = ctx + (size_t)b * (CDF_ * LL);

    v8f acc[2][2];
#pragma unroll
    for (int ms = 0; ms < 2; ++ms)
#pragma unroll
        for (int ns = 0; ns < 2; ++ns) zero8(acc[ms][ns]);

    for (int kk = 0; kk < 2; ++kk) {       // CDF in 2 chunks of 128
        __syncthreads();
        stage_pairs_128x64(sB, ctxb + (size_t)kk * 128 * LL, tid);
        __syncthreads();

        const float* A0 = W2 + (size_t)(m0 + l16) * CDF_ + kk * 128 + 2 * hf;
        const float* A1 = A0 + (size_t)16 * CDF_;
        for (int k0 = 0; k0 < 128; k0 += 4) {
            v2f a0 = *(const v2f*)(A0 + k0);
            v2f a1 = *(const v2f*)(A1 + k0);
            const int pb = ((k0 >> 1) + hf) * (STRP * 2);
            v2f b0 = *(const v2f*)&sB[pb + (n0 + l16) * 2];
            v2f b1 = *(const v2f*)&sB[pb + (n0 + 16 + l16) * 2];
            acc[0][0] = wmma_f32(a0, b0, acc[0][0]);
            acc[0][1] = wmma_f32(a0, b1, acc[0][1]);
            acc[1][0] = wmma_f32(a1, b0, acc[1][0]);
            acc[1][1] = wmma_f32(a1, b1, acc[1][1]);
        }
    }

    float* outb = sC + (size_t)b * HW_ * LL;
#pragma unroll
    for (int ms = 0; ms < 2; ++ms)
#pragma unroll
        for (int ns = 0; ns < 2; ++ns)
#pragma unroll
            for (int e = 0; e < 8; ++e) {
                int row = m0 + ms * 16 + e + 8 * hf;
                int col = n0 + ns * 16 + l16;
                outb[(size_t)row * LL + col] = acc[ms][ns][e];
            }
}

// ---------------------------------------------------------------------------
// k2: attn partial[b][ks][i][l] = sum_{k in slice} wC[b][i][k]*sC[b][k][l]
// grid (KSPLIT, B); K-slice = 512, staged in 4 LDS chunks of 128.
// ---------------------------------------------------------------------------
__global__ __launch_bounds__(256) void k2_attn_partial(const float* __restrict__ wc,
                                                       const float* __restrict__ sC,
                                                       float* __restrict__ part) {
    __shared__ float sB[64 * STRP * 2];
    const int b    = blockIdx.y;
    const int ks   = blockIdx.x;
    const int tid  = threadIdx.x;
    const int wave = tid >> 5;
    const int lane = tid & 31;
    const int hf   = lane >> 4;
    const int l16  = lane & 15;
    const int mg   = wave >> 1;
    const int nh   = wave & 1;
    const int m0   = mg * 32;              // IDF rows
    const int n0   = nh * 32;              // L cols
    const int kbase = ks * (HW_ / KSPLIT); // 512-wide K slice

    const float* sCb = sC + (size_t)b * HW_ * LL;

    v8f acc[2][2];
#pragma unroll
    for (int ms = 0; ms < 2; ++ms)
#pragma unroll
        for (int ns = 0; ns < 2; ++ns) zero8(acc[ms][ns]);

    for (int kk = 0; kk < 4; ++kk) {
        __syncthreads();
        stage_pairs_128x64(sB, sCb + (size_t)(kbase + kk * 128) * LL, tid);
        __syncthreads();

        const float* A0 = wc + (size_t)b * IDF_ * HW_ + (size_t)(m0 + l16) * HW_
                          + kbase + kk * 128 + 2 * hf;
        const float* A1 = A0 + (size_t)16 * HW_;
        for (int k0 = 0; k0 < 128; k0 += 4) {
            v2f a0 = *(const v2f*)(A0 + k0);
            v2f a1 = *(const v2f*)(A1 + k0);
            const int pb = ((k0 >> 1) + hf) * (STRP * 2);
            v2f b0 = *(const v2f*)&sB[pb + (n0 + l16) * 2];
            v2f b1 = *(const v2f*)&sB[pb + (n0 + 16 + l16) * 2];
            acc[0][0] = wmma_f32(a0, b0, acc[0][0]);
            acc[0][1] = wmma_f32(a0, b1, acc[0][1]);
            acc[1][0] = wmma_f32(a1, b0, acc[1][0]);
            acc[1][1] = wmma_f32(a1, b1, acc[1][1]);
        }
    }

    float* outp = part + ((size_t)(b * KSPLIT + ks)) * IDF_ * LL;
#pragma unroll
    for (int ms = 0; ms < 2; ++ms)
#pragma unroll
        for (int ns = 0; ns < 2; ++ns)
#pragma unroll
            for (int e = 0; e < 8; ++e) {
                int row = m0 + ms * 16 + e + 8 * hf;
                int col = n0 + ns * 16 + l16;
                outp[(size_t)row * LL + col] = acc[ms][ns][e];
            }
}

// ---------------------------------------------------------------------------
// k3: reduce KSPLIT partials + softmax over L (=64) per (b,i) row.
// One wave32 per row, 2 elements per lane; __shfl_xor reductions.
// ---------------------------------------------------------------------------
__global__ __launch_bounds__(256) void k3_softmax(const float* __restrict__ part,
                                                  float* __restrict__ attn,
                                                  float* __restrict__ attn_out) {
    const int tid  = threadIdx.x;
    const int wave = tid >> 5;
    const int lane = tid & 31;
    const int row  = blockIdx.x * 8 + wave;    // 0..4095
    const int b    = row >> 7;
    const int i    = row & 127;

    float v0 = 0.0f, v1 = 0.0f;
#pragma unroll
    for (int ks = 0; ks < KSPLIT; ++ks) {
        const float* p = part + (((size_t)(b * KSPLIT + ks)) * IDF_ + i) * LL;
        v0 += p[lane];
        v1 += p[lane + 32];
    }

    float m = fmaxf(v0, v1);
#pragma unroll
    for (int off = 16; off > 0; off >>= 1) m = fmaxf(m, __shfl_xor(m, off, 32));
    float e0 = __expf(v0 - m), e1 = __expf(v1 - m);
    float s = e0 + e1;
#pragma unroll
    for (int off = 16; off > 0; off >>= 1) s += __shfl_xor(s, off, 32);
    float rs = 1.0f / s;
    e0 *= rs;
    e1 *= rs;

    attn[(size_t)row * LL + lane]      = e0;
    attn[(size_t)row * LL + lane + 32] = e1;
    attn_out[((size_t)b * LL + lane) * IDF_ + i]      = e0;
    attn_out[((size_t)b * LL + lane + 32) * IDF_ + i] = e1;
}

// ---------------------------------------------------------------------------
// k4: out[b][i][o] = sum_l attn_sm[b][i][l] * sC[b][o][l]
//   = attn_sm(128x64) @ sC^T(64x4096).
// B = sC^T staged pair-interleaved: sB2[l/2][oo] = {sC[oo][l], sC[oo][l+1]}
// (contiguous float2 in global -> clean transpose). grid (HW/128, B).
// Wave (mg,nh): rows mg*32..+31, cols o0 + nh*64 .. +63.
// ---------------------------------------------------------------------------
__global__ __launch_bounds__(256) void k4_wc(const float* __restrict__ sC,
                                             const float* __restrict__ attn,
                                             float* __restrict__ out) {
    __shared__ float sB[32 * STRP2 * 2];   // 33 KB
    const int b    = blockIdx.y;
    const int o0   = blockIdx.x * 128;
    const int tid  = threadIdx.x;
    const int wave = tid >> 5;
    const int lane = tid & 31;
    const int hf   = lane >> 4;
    const int l16  = lane & 15;
    const int mg   = wave >> 1;
    const int nh   = wave & 1;
    const int m0   = mg * 32;

    const float* sCb = sC + (size_t)b * HW_ * LL;

    // stage: 128 oo-rows x 16 col-quads = 2048 units, 8 per thread
#pragma unroll
    for (int it = 0; it < 8; ++it) {
        int idx = tid + it * 256;
        int oo = idx >> 4;                 // 0..127
        int c  = idx & 15;                 // float4 within the 64-wide l row
        float4 v = *(const float4*)(sCb + (size_t)(o0 + oo) * LL + 4 * c);
        // pair-row p=2c   holds l = 4c,4c+1 -> {v.x, v.y}
        // pair-row p=2c+1 holds l = 4c+2,4c+3 -> {v.z, v.w}
        v2f s0; s0.x = v.x; s0.y = v.y;
        v2f s1; s1.x = v.z; s1.y = v.w;
        *(v2f*)&sB[((2 * c) * STRP2 + oo) * 2]     = s0;
        *(v2f*)&sB[((2 * c + 1) * STRP2 + oo) * 2] = s1;
    }
    __syncthreads();

    // preload A fragments: 2 M-strips x 4 k-steps (K=16)
    v2f a[2][4];
#pragma unroll
    for (int ms = 0; ms < 2; ++ms) {
        const float* Ar = attn + ((size_t)b * IDF_ + m0 + ms * 16 + l16) * LL + 2 * hf;
#pragma unroll
        for (int kt = 0; kt < 4; ++kt) a[ms][kt] = *(const v2f*)(Ar + kt * 4);
    }

    float* outb = out + (size_t)b * IDF_ * HW_;
#pragma unroll
    for (int nt = 0; nt < 4; ++nt) {
        const int oloc = nh * 64 + nt * 16 + l16;   // col within 128-wide chunk
        v8f acc0, acc1;
        zero8(acc0);
        zero8(acc1);
#pragma unroll
        for (int kt = 0; kt < 4; ++kt) {
            // k rows kt*4 + 2*hf, +1  ->  pair-row kt*2 + hf
            v2f bb = *(const v2f*)&sB[((kt * 2 + hf) * STRP2 + oloc) * 2];
            acc0 = wmma_f32(a[0][kt], bb, acc0);
            acc1 = wmma_f32(a[1][kt], bb, acc1);
        }
#pragma unroll
        for (int e = 0; e < 8; ++e) {
            int col = o0 + oloc;
            outb[(size_t)(m0 + e + 8 * hf) * HW_ + col]      = acc0[e];
            outb[(size_t)(m0 + 16 + e + 8 * hf) * HW_ + col] = acc1[e];
        }
    }
}

// ---------------------------------------------------------------------------
extern "C" void kernel_launch(void* const* d_in, const int* in_sizes, int n_in,
                              void* d_out, int out_size, void* d_ws, size_t ws_size,
                              hipStream_t stream) {
    const float* wC  = (const float*)d_in[0];  // [B, IDF, HW]
    const float* ctx = (const float*)d_in[1];  // [B, CDF, L]
    const float* W2  = (const float*)d_in[2];  // [HW, CDF]
    // d_in[3], d_in[4] are ih/iw scalars (compile-time constants here)

    float* out = (float*)d_out;                // wc [B,IDF,HW] then attn_c [B,L,IDF]

    // workspace layout (~43 MB):
    float* sC   = (float*)d_ws;                             // B*HW*L     = 33.5 MB
    float* part = sC + (size_t)B_ * HW_ * LL;               // B*KS*IDF*L =  8.4 MB
    float* attn = part + (size_t)B_ * KSPLIT * IDF_ * LL;   // B*IDF*L    =  1.0 MB

    k1_sourceC<<<dim3(HW_ / 128, B_), 256, 0, stream>>>(W2, ctx, sC);
    k2_attn_partial<<<dim3(KSPLIT, B_), 256, 0, stream>>>(wC, sC, part);
    k3_softmax<<<dim3((B_ * IDF_) / 8), 256, 0, stream>>>(
        part, attn, out + (size_t)B_ * IDF_ * HW_);
    k4_wc<<<dim3(HW_ / 128, B_), 256, 0, stream>>>(sC, attn, out);
}